// rpnProposalLayer_15934328668701
// MI455X (gfx1250) — compile-verified
//
#include <hip/hip_runtime.h>
#include <hip/hip_bf16.h>

// ---------------------------------------------------------------------------
// RPN proposal layer for MI455X (gfx1250, wave32, WGP with 320KB LDS).
//   B=16, A=261888 anchors, PRE_NMS=6000, POST_NMS=1000, NMS_THRESH=0.7
// Pipeline:
//   K1 select : per-batch exact radix-select of the 6000th score (two 16-bit
//               histogram passes in a 256KB LDS histogram; hierarchical
//               parallel suffix scan for the threshold bin)
//   K2 compact: atomic compaction of top-6000 (idx, score) per batch
//   K3 nms    : one workgroup per batch; decode+clip the 6000 boxes into LDS
//               (144KB resident), stage scores with the Tensor Data Mover
//               (tensor_load_to_lds + s_wait_tensorcnt), then 1000 iterations
//               of block-argmax + IoU suppression entirely in LDS.
// ---------------------------------------------------------------------------

#define BATCH     16
#define NANCH     261888
#define PRE_NMS   6000
#define POST_NMS  1000
#define NMS_TH    0.7f
#define NBINS     65536

// ---- anchor regeneration (matches numpy float64 _generate_anchors) ---------
__device__ __forceinline__ void anchor_coords(int a, float& x1, float& y1,
                                              float& x2, float& y2) {
  int base, fw, stride, scale;
  if (a < 196608)      { base = 0;      fw = 256; stride = 4;  scale = 4;  }
  else if (a < 245760) { base = 196608; fw = 128; stride = 8;  scale = 8;  }
  else if (a < 258048) { base = 245760; fw = 64;  stride = 16; scale = 16; }
  else if (a < 261120) { base = 258048; fw = 32;  stride = 32; scale = 32; }
  else                 { base = 261120; fw = 16;  stride = 64; scale = 64; }
  int idx = a - base;
  int r   = idx % 3;
  int loc = idx / 3;
  int col = loc % fw;
  int row = loc / fw;
  const double SQ[3] = {0.7071067811865476, 1.0, 1.4142135623730951};
  double sr = SQ[r];
  double ws = (double)scale * 8.0 * sr;
  double hs = (double)scale * 8.0 / sr;
  double cx = (col + 0.5) * (double)stride;
  double cy = (row + 0.5) * (double)stride;
  x1 = (float)(cx - 0.5 * ws);
  y1 = (float)(cy - 0.5 * hs);
  x2 = (float)(cx + 0.5 * ws);
  y2 = (float)(cy + 0.5 * hs);
}

// ---- hierarchical threshold finder over a 65536-bin LDS histogram ----------
// Finds bin T and above = count(keys in bins > T) s.t. above < target and
// above + hist[T] >= target. 1024 threads sum 64-bin chunks in parallel,
// then thread 0 scans 1024 chunk sums + <=64 bins (vs 65536 serial).
// Requires blockDim.x == 1024; leaves result in *s_bin / *s_above.
__device__ __forceinline__ void radix_find(const unsigned* __restrict__ hist,
                                           unsigned* __restrict__ chunkSum,
                                           unsigned target,
                                           unsigned* s_bin, unsigned* s_above) {
  const int c = threadIdx.x;
  unsigned s = 0;
#pragma unroll 8
  for (int i = 0; i < 64; ++i) s += hist[c * 64 + i];
  chunkSum[c] = s;
  __syncthreads();
  if (threadIdx.x == 0) {
    unsigned cum = 0; int cc = 1023;
    for (; cc > 0; --cc) {
      unsigned cs = chunkSum[cc];
      if (cum + cs >= target) break;
      cum += cs;
    }
    int bin = cc * 64 + 63;
    for (; bin > cc * 64; --bin) {
      unsigned h = hist[bin];
      if (cum + h >= target) break;
      cum += h;
    }
    *s_bin = (unsigned)bin; *s_above = cum;
  }
  __syncthreads();
}

// ---- K1: exact per-batch radix select of the PRE_NMS-th largest score ------
// Scores are fg probs in [0,1): positive floats, so uint bit ordering equals
// float ordering. Coalesced float2 loads (score = .y of each pair).
__global__ void __launch_bounds__(1024)
select_kernel(const float* __restrict__ probs,
              unsigned* __restrict__ keyThresh, int* __restrict__ nStrict,
              unsigned* __restrict__ cntS, unsigned* __restrict__ cntE) {
  extern __shared__ unsigned hist[];            // NBINS entries (256KB)
  __shared__ unsigned chunkSum[1024];           // 4KB scan scratch
  __shared__ unsigned s_coarse, s_above, s_low, s_lowAbove;
  const int b = blockIdx.x;
  const float2* sc2 = (const float2*)(probs + (size_t)b * NANCH * 2);

  for (int i = threadIdx.x; i < NBINS; i += blockDim.x) hist[i] = 0u;
  __syncthreads();
  for (int a = threadIdx.x; a < NANCH; a += blockDim.x) {
    if (a + 8192 < NANCH) __builtin_prefetch(&sc2[a + 8192], 0, 0);
    unsigned key = __float_as_uint(sc2[a].y);
    atomicAdd(&hist[key >> 16], 1u);
  }
  __syncthreads();
  radix_find(hist, chunkSum, PRE_NMS, &s_coarse, &s_above);
  const unsigned coarse = s_coarse;
  const unsigned above  = s_above;

  for (int i = threadIdx.x; i < NBINS; i += blockDim.x) hist[i] = 0u;
  __syncthreads();
  for (int a = threadIdx.x; a < NANCH; a += blockDim.x) {
    unsigned key = __float_as_uint(sc2[a].y);
    if ((key >> 16) == coarse) atomicAdd(&hist[key & 0xFFFFu], 1u);
  }
  __syncthreads();
  radix_find(hist, chunkSum, PRE_NMS - above, &s_low, &s_lowAbove);
  if (threadIdx.x == 0) {
    keyThresh[b] = (coarse << 16) | s_low;
    nStrict[b]   = (int)(above + s_lowAbove);     // keys strictly > threshold
    cntS[b] = 0u; cntE[b] = 0u;                   // reset compaction counters
  }
}

// ---- K2: compaction of top-6000 (index, score) per batch -------------------
__global__ void __launch_bounds__(256)
compact_kernel(const float* __restrict__ probs,
               const unsigned* __restrict__ keyThresh,
               const int* __restrict__ nStrict,
               unsigned* __restrict__ cntS, unsigned* __restrict__ cntE,
               int* __restrict__ top_idx, float* __restrict__ top_score) {
  const int b = blockIdx.y;
  const int a = blockIdx.x * blockDim.x + threadIdx.x;
  if (a >= NANCH) return;
  const float2* sc2 = (const float2*)(probs + (size_t)b * NANCH * 2);
  const unsigned key = __float_as_uint(sc2[a].y);
  const unsigned thr = keyThresh[b];
  if (key > thr) {
    unsigned p = atomicAdd(&cntS[b], 1u);        // p in [0, nStrict)
    top_idx[(size_t)b * PRE_NMS + p]   = a;
    top_score[(size_t)b * PRE_NMS + p] = __uint_as_float(key);
  } else if (key == thr) {
    unsigned p = atomicAdd(&cntE[b], 1u);
    unsigned slot = (unsigned)nStrict[b] + p;
    if (slot < PRE_NMS) {
      top_idx[(size_t)b * PRE_NMS + slot]   = a;
      top_score[(size_t)b * PRE_NMS + slot] = __uint_as_float(key);
    }
  }
}

// ---- K3: decode+clip into LDS, TDM score staging, greedy NMS, output -------
#define NMS_BT 1024
// 6 arrays of PRE_NMS floats + 32-wave reduction scratch
#define NMS_LDS_BYTES ((6 * PRE_NMS + 64 + 16) * 4)

__global__ void __launch_bounds__(NMS_BT)
nms_kernel(const float* __restrict__ deltas, const float* __restrict__ img_info,
           const int* __restrict__ top_idx, const float* __restrict__ top_score,
           float* __restrict__ out) {
  extern __shared__ float smem[];
  float* sx1 = smem;
  float* sy1 = sx1 + PRE_NMS;
  float* sx2 = sy1 + PRE_NMS;
  float* sy2 = sx2 + PRE_NMS;
  float* sar = sy2 + PRE_NMS;
  float* ssc = sar + PRE_NMS;
  float* redV = ssc + PRE_NMS;                 // 32 cross-wave partial values
  int*   redI = (int*)(redV + 32);             // 32 cross-wave partial indices
  __shared__ int   sj;
  __shared__ float sv;

  const int b = blockIdx.x;
  const unsigned t = threadIdx.x;
  const unsigned lane = t & 31u;               // wave32
  const unsigned wid  = t >> 5;                // 32 waves per block

  // --- Stage scores global->LDS with the Tensor Data Mover (wave 0 only) ---
#if defined(__gfx1250__)
  if (t < 32u) {
    typedef unsigned int u32x4 __attribute__((ext_vector_type(4)));
    typedef int i32x4 __attribute__((ext_vector_type(4)));
    typedef int i32x8 __attribute__((ext_vector_type(8)));
    unsigned lds_off = (unsigned)(unsigned long long)(void*)ssc;      // LDS byte offset
    unsigned long long ga =
        (unsigned long long)(const void*)(top_score + (size_t)b * PRE_NMS);
    u32x4 g0;
    g0[0] = 1u;                                        // count=1, user-mode D#
    g0[1] = lds_off;                                   // lds_addr
    g0[2] = (unsigned)(ga & 0xFFFFFFFFu);              // global_addr[31:0]
    g0[3] = (unsigned)((ga >> 32) & 0x1FFFFFFu) | (2u << 30);  // addr[56:32], type=2
    i32x8 g1;
    g1[0] = (2 << 16);                                 // data_size=4B, no multicast
    g1[1] = (int)(((unsigned)PRE_NMS & 0xFFFFu) << 16);        // tensor_dim0[15:0]
    g1[2] = (int)(((unsigned)PRE_NMS >> 16) & 0xFFFFu) | (1 << 16); // dim0 hi | dim1=1
    g1[3] = (int)(((unsigned)PRE_NMS) << 16);          // tile_dim0 = 6000
    g1[4] = 1;                                         // tile_dim1=1, tile_dim2=0
    g1[5] = PRE_NMS;                                   // tensor_dim0_stride lo32
    g1[6] = 0;
    g1[7] = 0;
    i32x4 z4; z4[0] = 0; z4[1] = 0; z4[2] = 0; z4[3] = 0;
#if __clang_major__ >= 23
    i32x8 z8; z8[0]=0; z8[1]=0; z8[2]=0; z8[3]=0; z8[4]=0; z8[5]=0; z8[6]=0; z8[7]=0;
    __builtin_amdgcn_tensor_load_to_lds(g0, g1, z4, z4, z8, 0);
#else
    __builtin_amdgcn_tensor_load_to_lds(g0, g1, z4, z4, 0);
#endif
    __builtin_amdgcn_s_wait_tensorcnt(0);
  }
#endif
  __syncthreads();   // TDM complete before any other writes to ssc

  // --- Decode + clip selected boxes directly into LDS (144KB resident) -----
  const float hmax = img_info[b * 3 + 0] - 1.0f;
  const float wmax = img_info[b * 3 + 1] - 1.0f;
  for (int k = (int)t; k < PRE_NMS; k += NMS_BT) {
    const int a = top_idx[(size_t)b * PRE_NMS + k];
    float ax1, ay1, ax2, ay2;
    anchor_coords(a, ax1, ay1, ax2, ay2);
    const float w  = ax2 - ax1 + 1.0f;
    const float h  = ay2 - ay1 + 1.0f;
    const float cx = ax1 + 0.5f * w;
    const float cy = ay1 + 0.5f * h;
    const float* d = deltas + ((size_t)b * NANCH + a) * 4;
    const float pcx = d[0] * w + cx;
    const float pcy = d[1] * h + cy;
    const float pw  = expf(d[2]) * w;
    const float ph  = expf(d[3]) * h;
    float x1 = fminf(fmaxf(pcx - 0.5f * pw, 0.0f), wmax);
    float y1 = fminf(fmaxf(pcy - 0.5f * ph, 0.0f), hmax);
    float x2 = fminf(fmaxf(pcx + 0.5f * pw, 0.0f), wmax);
    float y2 = fminf(fmaxf(pcy + 0.5f * ph, 0.0f), hmax);
    sx1[k] = x1; sy1[k] = y1; sx2[k] = x2; sy2[k] = y2;
    sar[k] = (x2 - x1 + 1.0f) * (y2 - y1 + 1.0f);
    ssc[k] = top_score[(size_t)b * PRE_NMS + k];  // deterministic overwrite of TDM data
  }
  __syncthreads();

  // --- Greedy NMS: 1000 x (block argmax -> record -> IoU suppression) ------
  for (int it = 0; it < POST_NMS; ++it) {
    float bv = -1e30f; int bi = 0;
    for (int k = (int)t; k < PRE_NMS; k += NMS_BT) {
      float v = ssc[k];
      if (v > bv) { bv = v; bi = k; }
    }
    // wave32 reduction (value, first-index tie-break)
    for (int off = 16; off > 0; off >>= 1) {
      float ov = __shfl_down(bv, off, 32);
      int   oi = __shfl_down(bi, off, 32);
      if (ov > bv || (ov == bv && oi < bi)) { bv = ov; bi = oi; }
    }
    if (lane == 0) { redV[wid] = bv; redI[wid] = bi; }
    __syncthreads();
    if (wid == 0) {
      bv = redV[lane]; bi = redI[lane];
      for (int off = 16; off > 0; off >>= 1) {
        float ov = __shfl_down(bv, off, 32);
        int   oi = __shfl_down(bi, off, 32);
        if (ov > bv || (ov == bv && oi < bi)) { bv = ov; bi = oi; }
      }
      if (lane == 0) { sj = bi; sv = bv; }
    }
    __syncthreads();
    const int   j  = sj;
    const float vj = sv;
    const float jx1 = sx1[j], jy1 = sy1[j], jx2 = sx2[j], jy2 = sy2[j];
    const float jar = sar[j];
    if (t == 0) {
      const float m = (vj > 0.0f) ? 1.0f : 0.0f;
      float* o = out + ((size_t)b * POST_NMS + it) * 5;
      o[0] = (float)b;
      o[1] = jx1 * m; o[2] = jy1 * m; o[3] = jx2 * m; o[4] = jy2 * m;
    }
    for (int k = (int)t; k < PRE_NMS; k += NMS_BT) {
      const float xx1 = fmaxf(sx1[k], jx1);
      const float yy1 = fmaxf(sy1[k], jy1);
      const float xx2 = fminf(sx2[k], jx2);
      const float yy2 = fminf(sy2[k], jy2);
      const float iw = fmaxf(xx2 - xx1 + 1.0f, 0.0f);
      const float ih = fmaxf(yy2 - yy1 + 1.0f, 0.0f);
      const float inter = iw * ih;
      const float iou = inter / (sar[k] + jar - inter);
      if (iou > NMS_TH) ssc[k] = -1.0f;
    }
    __syncthreads();
  }
}

// ---------------------------------------------------------------------------
extern "C" void kernel_launch(void* const* d_in, const int* in_sizes, int n_in,
                              void* d_out, int out_size, void* d_ws, size_t ws_size,
                              hipStream_t stream) {
  (void)in_sizes; (void)n_in; (void)out_size; (void)ws_size;
  const float* probs    = (const float*)d_in[0];   // [B, A, 2]
  const float* deltas   = (const float*)d_in[1];   // [B, A, 4]
  const float* img_info = (const float*)d_in[2];   // [B, 3]
  float* out = (float*)d_out;                      // [B, 1000, 5]

  char* ws = (char*)d_ws;
  unsigned* keyThresh = (unsigned*)(ws + 0);
  int*      nStrict   = (int*)(ws + 64);
  unsigned* cntS      = (unsigned*)(ws + 128);
  unsigned* cntE      = (unsigned*)(ws + 192);
  int*      top_idx   = (int*)(ws + 256);
  float*    top_score = (float*)(ws + 256 + (size_t)BATCH * PRE_NMS * 4);

  // Opt in to large dynamic LDS (CDNA5 WGP has 320KB).
  (void)hipFuncSetAttribute((const void*)select_kernel,
                            hipFuncAttributeMaxDynamicSharedMemorySize,
                            NBINS * 4);
  (void)hipFuncSetAttribute((const void*)nms_kernel,
                            hipFuncAttributeMaxDynamicSharedMemorySize,
                            NMS_LDS_BYTES);

  select_kernel<<<BATCH, 1024, NBINS * 4, stream>>>(probs, keyThresh, nStrict,
                                                    cntS, cntE);
  dim3 cg((NANCH + 255) / 256, BATCH);
  compact_kernel<<<cg, 256, 0, stream>>>(probs, keyThresh, nStrict, cntS, cntE,
                                         top_idx, top_score);
  nms_kernel<<<BATCH, NMS_BT, NMS_LDS_BYTES, stream>>>(deltas, img_info,
                                                       top_idx, top_score, out);
}